// ConstLoss_63282048139751
// MI455X (gfx1250) — compile-verified
//
#include <hip/hip_runtime.h>

typedef float v2f __attribute__((ext_vector_type(2)));
typedef float v8f __attribute__((ext_vector_type(8)));

#define NROWS 8192
#define DDIM  128
#define TILE  128
#define SROW  132   // 128 + 4 pad floats: column-slice readers hit 16 distinct LDS banks
#define EPSN  1e-8f

#define WMMA_F32(A, B, C) \
    __builtin_amdgcn_wmma_f32_16x16x4_f32(false, (A), false, (B), (short)0, (C), false, false)

__global__ void init_acc_kernel(double* acc) { *acc = 0.0; }

// One wave (32 lanes) per row; rows [0,8192) -> Q, [8192,16384) -> K.
__global__ __launch_bounds__(256)
void normalize_rows_kernel(const float* __restrict__ q,
                           const float* __restrict__ k,
                           float* __restrict__ qn,
                           float* __restrict__ kn) {
    const int wave = threadIdx.x >> 5;
    const int lane = threadIdx.x & 31;
    const int row  = blockIdx.x * 8 + wave;            // 0..16383

    const float* src;
    float* dst;
    if (row < NROWS) {
        src = q  + (size_t)row * DDIM;
        dst = qn + (size_t)row * DDIM;
    } else {
        src = k  + (size_t)(row - NROWS) * DDIM;
        dst = kn + (size_t)(row - NROWS) * DDIM;
    }

    float4 v = ((const float4*)src)[lane];             // 32 lanes x 4 = 128 cols
    float s = v.x * v.x + v.y * v.y + v.z * v.z + v.w * v.w;
#pragma unroll
    for (int off = 16; off > 0; off >>= 1) s += __shfl_xor(s, off, 32);

    const float inv = 1.0f / fmaxf(sqrtf(s), EPSN);
    float4 o;
    o.x = v.x * inv; o.y = v.y * inv; o.z = v.z * inv; o.w = v.w * inv;
    ((float4*)dst)[lane] = o;
}

// Fused symmetric Gram-difference reduction.
// Grid 64x64 macro-tiles of 128x128; only tj >= ti computed (exact fp symmetry),
// off-diagonal tiles weighted 2x. Block = 512 threads = 16 waves in a 4x4 grid;
// each wave register-blocks a 32x32 quadrant: 2 A-frags x 2 B-frags -> 8 WMMAs
// per 8 LDS fragment loads, for both the Q and K accumulation chains.
__global__ __launch_bounds__(512)
void gram_diff_kernel(const float* __restrict__ qn,
                      const float* __restrict__ kn,
                      double* __restrict__ acc) {
    const int ti = blockIdx.x;
    const int tj = blockIdx.y;
    if (tj < ti) return;                               // upper-triangular tile set only
    const bool diag = (ti == tj);

    extern __shared__ float smem[];
    float* sQi = smem;                                 // TILE x SROW
    float* sKi = sQi + TILE * SROW;
    float* sQj = diag ? sQi : (sKi + TILE * SROW);
    float* sKj = diag ? sKi : (sKi + 2 * TILE * SROW);
    __shared__ float red[16];

    const int i0 = ti * TILE;
    const int j0 = tj * TILE;

    // Cooperative staging: 128 rows x 32 float4 per panel = 4096 chunks, 512 thr -> 8 each.
    {
        const float4* gQi = (const float4*)(qn + (size_t)i0 * DDIM);
        const float4* gKi = (const float4*)(kn + (size_t)i0 * DDIM);
        const float4* gQj = (const float4*)(qn + (size_t)j0 * DDIM);
        const float4* gKj = (const float4*)(kn + (size_t)j0 * DDIM);
#pragma unroll
        for (int t = 0; t < 8; ++t) {
            const int idx = threadIdx.x + t * 512;     // 0..4095
            const int r   = idx >> 5;                  // row 0..127
            const int c4  = idx & 31;                  // float4 col 0..31
            const int so  = ((r * SROW) >> 2) + c4;    // SROW%4==0 -> float4-aligned
            ((float4*)sQi)[so] = gQi[idx];
            ((float4*)sKi)[so] = gKi[idx];
            if (!diag) {
                ((float4*)sQj)[so] = gQj[idx];
                ((float4*)sKj)[so] = gKj[idx];
            }
        }
    }
    __syncthreads();

    const int wave = threadIdx.x >> 5;
    const int lane = threadIdx.x & 31;
    const int wi   = (wave >> 2) * 32;                 // quadrant row origin
    const int wj   = (wave & 3) * 32;                  // quadrant col origin

    // f32 16x4 A-fragment: lanes 0-15 -> M=lane, K=0,1; lanes 16-31 -> M=lane-16, K=2,3.
    // B (4x16) of A*A^T uses identical row-major addressing (row = N).
    const int m  = lane & 15;
    const int kb = (lane >> 4) * 2;

    const float* aQ0 = sQi + (wi + m) * SROW + kb;
    const float* aQ1 = aQ0 + 16 * SROW;
    const float* aK0 = sKi + (wi + m) * SROW + kb;
    const float* aK1 = aK0 + 16 * SROW;
    const float* bQ0 = sQj + (wj + m) * SROW + kb;
    const float* bQ1 = bQ0 + 16 * SROW;
    const float* bK0 = sKj + (wj + m) * SROW + kb;
    const float* bK1 = bK0 + 16 * SROW;

    v8f cq00 = {}, cq01 = {}, cq10 = {}, cq11 = {};
    v8f ck00 = {}, ck01 = {}, ck10 = {}, ck11 = {};
#pragma unroll
    for (int kk = 0; kk < DDIM; kk += 4) {
        v2f aq0 = { aQ0[kk], aQ0[kk + 1] };
        v2f aq1 = { aQ1[kk], aQ1[kk + 1] };
        v2f bq0 = { bQ0[kk], bQ0[kk + 1] };
        v2f bq1 = { bQ1[kk], bQ1[kk + 1] };
        v2f ak0 = { aK0[kk], aK0[kk + 1] };
        v2f ak1 = { aK1[kk], aK1[kk + 1] };
        v2f bk0 = { bK0[kk], bK0[kk + 1] };
        v2f bk1 = { bK1[kk], bK1[kk + 1] };
        cq00 = WMMA_F32(aq0, bq0, cq00);
        cq01 = WMMA_F32(aq0, bq1, cq01);
        cq10 = WMMA_F32(aq1, bq0, cq10);
        cq11 = WMMA_F32(aq1, bq1, cq11);
        ck00 = WMMA_F32(ak0, bk0, ck00);
        ck01 = WMMA_F32(ak0, bk1, ck01);
        ck10 = WMMA_F32(ak1, bk0, ck10);
        ck11 = WMMA_F32(ak1, bk1, ck11);
    }

    // C/D layout: N = lane&15 (both halves), M = vgpr + 8*(lane>=16).
    const int Mb = (lane >> 4) * 8;
    float s = 0.0f;
    {
        const v8f* qs[4] = { &cq00, &cq01, &cq10, &cq11 };
        const v8f* ks[4] = { &ck00, &ck01, &ck10, &ck11 };
#pragma unroll
        for (int sub = 0; sub < 4; ++sub) {
            const int r   = sub >> 1;                  // A sub-tile index
            const int c   = sub & 1;                   // B sub-tile index
            const int gj  = j0 + wj + c * 16 + m;
            const int gi0 = i0 + wi + r * 16 + Mb;
            const v8f& qv = *qs[sub];
            const v8f& kv = *ks[sub];
#pragma unroll
            for (int v = 0; v < 8; ++v) {
                const float d = qv[v] - kv[v];
                s += (gi0 + v == gj) ? 0.0f : d * d;   // mask diagonal i==j
            }
        }
    }

#pragma unroll
    for (int off = 16; off > 0; off >>= 1) s += __shfl_xor(s, off, 32);
    if (lane == 0) red[wave] = s;
    __syncthreads();
    if (threadIdx.x == 0) {
        float tot = 0.0f;
#pragma unroll
        for (int w = 0; w < 16; ++w) tot += red[w];
        const double weight = diag ? 1.0 : 2.0;        // exact: Gram diff^2 is fp-symmetric
        atomicAdd(acc, weight * (double)tot);
    }
}

__global__ void finalize_kernel(const double* __restrict__ acc,
                                float* __restrict__ out) {
    out[0] = (float)(*acc / ((double)NROWS * (double)(NROWS - 1)));
}

extern "C" void kernel_launch(void* const* d_in, const int* in_sizes, int n_in,
                              void* d_out, int out_size, void* d_ws, size_t ws_size,
                              hipStream_t stream) {
    const float* q = (const float*)d_in[0];
    const float* k = (const float*)d_in[1];

    float*  qn  = (float*)d_ws;
    float*  kn  = qn + (size_t)NROWS * DDIM;
    double* acc = (double*)((char*)d_ws + 2ull * NROWS * DDIM * sizeof(float));

    init_acc_kernel<<<1, 1, 0, stream>>>(acc);
    normalize_rows_kernel<<<(2 * NROWS) / 8, 256, 0, stream>>>(q, k, qn, kn);

    dim3 grid(NROWS / TILE, NROWS / TILE);             // 64 x 64 macro-tiles (upper tri live)
    const size_t lds_bytes = 4ull * TILE * SROW * sizeof(float);  // ~264 KB (<= 320 KB/WGP)
    gram_diff_kernel<<<grid, 512, lds_bytes, stream>>>(qn, kn, acc);

    finalize_kernel<<<1, 1, 0, stream>>>(acc, (float*)d_out);
}